// VQ_46110768890166
// MI455X (gfx1250) — compile-verified
//
#include <hip/hip_runtime.h>

typedef float v2f __attribute__((ext_vector_type(2)));
typedef float v4f __attribute__((ext_vector_type(4)));
typedef float v8f __attribute__((ext_vector_type(8)));

#define ROWS   32768      // 8 * 4096
#define DIMN   256
#define KCODES 1024
#define QCOUNT (ROWS * DIMN)   // 8388608
#define ZPAD   260             // 256 + 4 dword pad: conflict-free b64 A-reads
#define GATHER_BLOCKS 4096

// ---------------------------------------------------------------------------
// Kernel P: c[n] = ||emb[n]||^2 for all 1024 codes.
// ---------------------------------------------------------------------------
__global__ void __launch_bounds__(256) vq_prep(const float* __restrict__ emb,
                                               float* __restrict__ c) {
    const int n = blockIdx.x * blockDim.x + threadIdx.x;
    if (n < KCODES) {
        const v4f* e = reinterpret_cast<const v4f*>(emb + (size_t)n * DIMN);
        float s = 0.0f;
#pragma unroll 8
        for (int i = 0; i < DIMN / 4; ++i) {
            v4f v = e[i];
            s += v[0] * v[0] + v[1] * v[1] + v[2] * v[2] + v[3] * v[3];
        }
        c[n] = s;
    }
}

// ---------------------------------------------------------------------------
// Kernel A: z = x @ W^T + b   via V_WMMA_F32_16X16X4_F32.
// One wave computes a 16(M) x 64(N) slab, K = 256 in steps of 4.
// A layout (f32 16x4): lane l holds M=l; v0/v1 = K = kOff, kOff+1 (kOff=2*half)
// B layout (f32 4x16): lane l holds N=l; v0/v1 = K = kOff, kOff+1
// C/D layout: VGPR r -> row (r + 8*half), col = lane&15
// ---------------------------------------------------------------------------
__global__ void __launch_bounds__(256) vq_proj(const float* __restrict__ x,
                                               const float* __restrict__ W,
                                               const float* __restrict__ b,
                                               float* __restrict__ z) {
    const int lane = threadIdx.x & 31;
    const int half = lane >> 4;
    const int l    = lane & 15;
    const int wave = blockIdx.x * (blockDim.x >> 5) + (threadIdx.x >> 5);
    const int mTile = wave >> 2;        // 0..2047
    const int nSlab = wave & 3;         // 0..3
    const int mBase = mTile * 16;
    const int nBase = nSlab * 64;
    const int kOff  = half * 2;

    const float* xRow = x + (size_t)(mBase + l) * DIMN;
    const float* w0 = W + (size_t)(nBase +  0 + l) * DIMN;
    const float* w1 = W + (size_t)(nBase + 16 + l) * DIMN;
    const float* w2 = W + (size_t)(nBase + 32 + l) * DIMN;
    const float* w3 = W + (size_t)(nBase + 48 + l) * DIMN;

    v8f a0 = {}, a1 = {}, a2 = {}, a3 = {};

#pragma unroll 4
    for (int k = 0; k < DIMN; k += 4) {
        v2f av = *reinterpret_cast<const v2f*>(xRow + k + kOff);
        v2f b0 = *reinterpret_cast<const v2f*>(w0 + k + kOff);
        v2f b1 = *reinterpret_cast<const v2f*>(w1 + k + kOff);
        v2f b2 = *reinterpret_cast<const v2f*>(w2 + k + kOff);
        v2f b3 = *reinterpret_cast<const v2f*>(w3 + k + kOff);
        a0 = __builtin_amdgcn_wmma_f32_16x16x4_f32(false, av, false, b0, (short)0, a0, false, false);
        a1 = __builtin_amdgcn_wmma_f32_16x16x4_f32(false, av, false, b1, (short)0, a1, false, false);
        a2 = __builtin_amdgcn_wmma_f32_16x16x4_f32(false, av, false, b2, (short)0, a2, false, false);
        a3 = __builtin_amdgcn_wmma_f32_16x16x4_f32(false, av, false, b3, (short)0, a3, false, false);
    }

    const float bias0 = b[nBase +  0 + l];
    const float bias1 = b[nBase + 16 + l];
    const float bias2 = b[nBase + 32 + l];
    const float bias3 = b[nBase + 48 + l];

#pragma unroll
    for (int r = 0; r < 8; ++r) {
        const int row = mBase + r + 8 * half;
        float* zr = z + (size_t)row * DIMN + nBase;
        zr[ 0 + l] = a0[r] + bias0;
        zr[16 + l] = a1[r] + bias1;
        zr[32 + l] = a2[r] + bias2;
        zr[48 + l] = a3[r] + bias3;
    }
}

// ---------------------------------------------------------------------------
// Kernel B: scores s = z @ emb^T (WMMA f32), fused argmin_n (c[n] - 2*s).
// One wave owns 16 rows. Rows 0..7 (8x256) of the z strip are staged once into LDS
// (row-major, stride ZPAD=260 dwords -> conflict-free b64 A-operand reads),
// then 16 slabs of 64 codes are streamed from global (emb is L2-resident).
// 2 waves per block -> 33 KB static LDS.
// ---------------------------------------------------------------------------
__global__ void __launch_bounds__(64) vq_argmin(const float* __restrict__ z,
                                                const float* __restrict__ emb,
                                                const float* __restrict__ c,
                                                int* __restrict__ idxOut,
                                                float* __restrict__ idxOutF) {
    __shared__ float zs[2 * 16 * ZPAD];

    const int lane = threadIdx.x & 31;
    const int half = lane >> 4;
    const int l    = lane & 15;
    const int w    = threadIdx.x >> 5;                 // 0..1
    const int wave = blockIdx.x * 2 + w;               // 0..2047
    const int mBase = wave * 16;
    const int kOff  = half * 2;

    float* zw = zs + w * 16 * ZPAD;

    // Stage rows 0..7 (8x256) of this wave's z strip into LDS (coalesced 512B global reads).
    const float* strip = z + (size_t)mBase * DIMN;
#pragma unroll
    for (int j = 0; j < 16; ++j) {
        const int r  = j >> 1;
        const int kf = (j & 1) * 128 + lane * 4;
        v4f val = *reinterpret_cast<const v4f*>(strip + j * 128 + lane * 4);
        *reinterpret_cast<v4f*>(zw + r * ZPAD + kf) = val;
    }
    __syncthreads();

    const float* aRow = zw + l * ZPAD;   // A operand: M = l

    float minv[8];
    int   mini[8];
#pragma unroll
    for (int r = 0; r < 8; ++r) { minv[r] = __builtin_huge_valf(); mini[r] = 0; }

    for (int slab = 0; slab < 16; ++slab) {
        const int nBase = slab * 64;
        const float* ePtr[4];
#pragma unroll
        for (int t = 0; t < 4; ++t)
            ePtr[t] = emb + (size_t)(nBase + t * 16 + l) * DIMN;

        v8f acc[4];
        const v8f vz = {};
#pragma unroll
        for (int t = 0; t < 4; ++t) acc[t] = vz;

#pragma unroll 4
        for (int k = 0; k < DIMN; k += 4) {
            v2f av = *reinterpret_cast<const v2f*>(aRow + k + kOff);   // LDS b64, conflict-free
#pragma unroll
            for (int t = 0; t < 4; ++t) {
                v2f bv = *reinterpret_cast<const v2f*>(ePtr[t] + k + kOff);
                acc[t] = __builtin_amdgcn_wmma_f32_16x16x4_f32(false, av, false, bv,
                                                               (short)0, acc[t], false, false);
            }
        }

#pragma unroll
        for (int t = 0; t < 4; ++t) {
            const int n  = nBase + t * 16 + l;
            const float cn = c[n];
#pragma unroll
            for (int r = 0; r < 8; ++r) {
                const float v = cn - 2.0f * acc[t][r];
                if (v < minv[r] || (v == minv[r] && n < mini[r])) { minv[r] = v; mini[r] = n; }
            }
        }
    }

    // Cross-lane argmin within each half-wave (16 lanes hold 16 different codes)
#pragma unroll
    for (int off = 1; off < 16; off <<= 1) {
#pragma unroll
        for (int r = 0; r < 8; ++r) {
            const float ov = __shfl_xor(minv[r], off, 32);
            const int   oi = __shfl_xor(mini[r], off, 32);
            if (ov < minv[r] || (ov == minv[r] && oi < mini[r])) { minv[r] = ov; mini[r] = oi; }
        }
    }

    if (l == 0) {
#pragma unroll
        for (int r = 0; r < 8; ++r) {
            const int row = mBase + r + 8 * half;
            idxOut[row]  = mini[r];
            idxOutF[row] = (float)mini[r];
        }
    }
}

// ---------------------------------------------------------------------------
// Kernel C: quant = emb[idx]; per-block partial of sum((emb[idx] - z)^2).
// One wave per row (256 floats -> 8 per lane as 2x float4). Deterministic:
// block partials go to ws, reduced in fixed order by vq_loss_reduce.
// ---------------------------------------------------------------------------
__global__ void __launch_bounds__(256) vq_gather(const float* __restrict__ emb,
                                                 const float* __restrict__ z,
                                                 const int* __restrict__ idx,
                                                 float* __restrict__ quant,
                                                 float* __restrict__ blockSums) {
    __shared__ float partial[8];
    const int lane = threadIdx.x & 31;
    const int w    = threadIdx.x >> 5;
    const int row  = blockIdx.x * 8 + w;

    const int code = idx[row];
    const float* e  = emb + (size_t)code * DIMN;
    const float* zr = z   + (size_t)row  * DIMN;
    float*       q  = quant + (size_t)row * DIMN;

    float s = 0.0f;
#pragma unroll
    for (int part = 0; part < 2; ++part) {
        const int off = part * 128 + lane * 4;
        v4f ev = *reinterpret_cast<const v4f*>(e + off);
        v4f zv = *reinterpret_cast<const v4f*>(zr + off);
        *reinterpret_cast<v4f*>(q + off) = ev;
        v4f d = ev - zv;
        s += d[0] * d[0] + d[1] * d[1] + d[2] * d[2] + d[3] * d[3];
    }
#pragma unroll
    for (int off = 16; off >= 1; off >>= 1) s += __shfl_xor(s, off, 32);
    if (lane == 0) partial[w] = s;
    __syncthreads();
    if (threadIdx.x == 0) {
        float t = 0.0f;
#pragma unroll
        for (int i = 0; i < 8; ++i) t += partial[i];
        blockSums[blockIdx.x] = t;
    }
}

// ---------------------------------------------------------------------------
// Kernel D: deterministic final loss reduction over 4096 block partials.
// loss = 1.25 * sum / 8388608   (BETA*mean + mean, forward values identical)
// ---------------------------------------------------------------------------
__global__ void __launch_bounds__(256) vq_loss_reduce(const float* __restrict__ blockSums,
                                                      float* __restrict__ lossSlot) {
    __shared__ float sh[256];
    float s = 0.0f;
#pragma unroll
    for (int j = 0; j < GATHER_BLOCKS / 256; ++j)
        s += blockSums[threadIdx.x + j * 256];
    sh[threadIdx.x] = s;
    __syncthreads();
    for (int off = 128; off > 0; off >>= 1) {
        if (threadIdx.x < off) sh[threadIdx.x] += sh[threadIdx.x + off];
        __syncthreads();
    }
    if (threadIdx.x == 0) lossSlot[0] = sh[0] * (1.25f / 8388608.0f);
}

// ---------------------------------------------------------------------------
// Host launcher
// ---------------------------------------------------------------------------
extern "C" void kernel_launch(void* const* d_in, const int* in_sizes, int n_in,
                              void* d_out, int out_size, void* d_ws, size_t ws_size,
                              hipStream_t stream) {
    (void)in_sizes; (void)n_in; (void)out_size; (void)ws_size;

    const float* x   = (const float*)d_in[0];   // [8,4096,256]
    const float* W   = (const float*)d_in[1];   // [256,256]
    const float* b   = (const float*)d_in[2];   // [256]
    const float* emb = (const float*)d_in[3];   // [1024,256]

    float* out      = (float*)d_out;
    float* quant    = out;                        // 8388608 floats
    float* idxF     = out + QCOUNT;               // 32768 floats (argmin indices)
    float* lossSlot = out + QCOUNT + ROWS;        // 1 float

    float* z     = (float*)d_ws;                  // 32768*256 floats (32 MB)
    float* cbuf  = z + (size_t)ROWS * DIMN;       // 1024 floats
    int*   idxI  = (int*)(cbuf + KCODES);         // 32768 ints
    float* bsums = (float*)(idxI + ROWS);         // 4096 floats

    vq_prep       <<<   4, 256, 0, stream>>>(emb, cbuf);
    vq_proj       <<<1024, 256, 0, stream>>>(x, W, b, z);        // 8192 waves
    vq_argmin     <<<1024,  64, 0, stream>>>(z, emb, cbuf, idxI, idxF); // 2048 waves
    vq_gather     <<<GATHER_BLOCKS, 256, 0, stream>>>(emb, z, idxI, quant, bsums);
    vq_loss_reduce<<<   1, 256, 0, stream>>>(bsums, lossSlot);
}